// Ibex_FANN_15049565405380
// MI455X (gfx1250) — compile-verified
//
#include <hip/hip_runtime.h>

typedef __attribute__((ext_vector_type(2))) float v2f;
typedef __attribute__((ext_vector_type(8))) float v8f;

#define K1 117      // layer-1 inner dim
#define N1 20       // layer-1 outputs
#define NITER 30    // ceil(117/4) k-steps of the 16x16x4 WMMA
#define WAVES 8
#define ROWS_PER_BLOCK 256   // 8 waves x 2 M-tiles x 16 rows
#define THREADS 256

__device__ __forceinline__ v2f load_a_frag(const float* __restrict__ xrow,
                                           int it, int koff) {
    const int k0 = it * 4 + koff;
    v2f a;
    if (k0 + 1 < K1) {                    // uniform for it < 29
        a[0] = xrow[it * 4];
        a[1] = xrow[it * 4 + 1];
    } else {                              // K tail (it == 29)
        a[0] = (k0 < K1) ? xrow[it * 4] : 0.f;
        a[1] = 0.f;
    }
    return a;
}

__device__ __forceinline__ v8f wmma4(v2f a, v2f b, v8f c) {
    return __builtin_amdgcn_wmma_f32_16x16x4_f32(
        false, a, false, b, (short)0, c, false, false);
}

__global__ __launch_bounds__(THREADS)
void fann_mlp_wmma_kernel(const float* __restrict__ X,
                          const float* __restrict__ W1,
                          const float* __restrict__ b1,
                          const float* __restrict__ W2,
                          const float* __restrict__ b2,
                          float* __restrict__ out,
                          int nTiles) {
    // B-fragments of W1^T, padded K:117->120, N:20->32, in WMMA lane order.
    // Entry e = it*32 + lane: .xy = tile0 (cols 0..15), .zw = tile1 (cols 16..31)
    __shared__ float4 bfrag[NITER * 32];          // 15,360 B
    __shared__ float  hbuf[WAVES][32][N1];        // 20,480 B

    const int tid  = threadIdx.x;
    const int lane = tid & 31;
    const int wave = tid >> 5;

    // ---- one-time prep: build W1^T fragments in LDS ----
    for (int e = tid; e < NITER * 32; e += THREADS) {
        int it = e >> 5;
        int l  = e & 31;
        int n  = l & 15;
        int kr = it * 4 + ((l >> 4) << 1);        // K row for this lane half
        float4 v;
        v.x = (kr     < K1)                 ? W1[n * K1 + kr]            : 0.f;
        v.y = (kr + 1 < K1)                 ? W1[n * K1 + kr + 1]        : 0.f;
        v.z = (kr     < K1 && (n+16) < N1)  ? W1[(n+16) * K1 + kr]       : 0.f;
        v.w = (kr + 1 < K1 && (n+16) < N1)  ? W1[(n+16) * K1 + kr + 1]   : 0.f;
        bfrag[e] = v;
    }
    __syncthreads();

    const int laneRow = lane & 15;                // A-matrix M index
    const int koff    = (lane >> 4) << 1;         // 0 or 2 (A-matrix K sublane)

    for (int tile = blockIdx.x; tile < nTiles; tile += gridDim.x) {
        const int rowBase = tile * ROWS_PER_BLOCK;
        const int rowG0   = rowBase + wave * 32 + laneRow;   // M-tile 0
        const int rowG1   = rowG0 + 16;                      // M-tile 1
        const float* __restrict__ xrow0 = X + (size_t)rowG0 * K1 + koff;
        const float* __restrict__ xrow1 = X + (size_t)rowG1 * K1 + koff;

        // prefetch toward this block's next grid-stride tile
        if (tile + (int)gridDim.x < nTiles) {
            __builtin_prefetch(X + ((size_t)rowG0 +
                               (size_t)gridDim.x * ROWS_PER_BLOCK) * K1, 0, 3);
        }

        v8f acc00 = {}, acc01 = {};   // rows 0..15  x cols {0..15, 16..31}
        v8f acc10 = {}, acc11 = {};   // rows 16..31 x cols {0..15, 16..31}

        // software pipeline: fragments for it are fetched at it-1
        v2f a0 = load_a_frag(xrow0, 0, koff);
        v2f a1 = load_a_frag(xrow1, 0, koff);
        float4 bv = bfrag[lane];

        #pragma unroll
        for (int it = 0; it < NITER; ++it) {
            v2f a0n = {}, a1n = {};
            float4 bvn = {};
            if (it + 1 < NITER) {
                a0n = load_a_frag(xrow0, it + 1, koff);
                a1n = load_a_frag(xrow1, it + 1, koff);
                bvn = bfrag[(it + 1) * 32 + lane];
            }
            v2f b0;  b0[0]  = bv.x; b0[1]  = bv.y;
            v2f b1f; b1f[0] = bv.z; b1f[1] = bv.w;
            acc00 = wmma4(a0, b0,  acc00);
            acc01 = wmma4(a0, b1f, acc01);
            acc10 = wmma4(a1, b0,  acc10);
            acc11 = wmma4(a1, b1f, acc11);
            a0 = a0n; a1 = a1n; bv = bvn;
        }

        // ---- layer-1 bias + requant (x1565, +2^14, trunc, >>15, clamp 0..255) ----
        const int mBase = (lane >> 4) << 3;       // C/D layout: lanes>=16 hold M=8..15
        const int c0 = lane & 15;
        const int c1 = 16 + c0;
        const float bc0 = b1[c0];
        const float bc1 = (c1 < N1) ? b1[c1] : 0.f;
        #pragma unroll
        for (int r = 0; r < 8; ++r) {
            const int m = mBase + r;
            // M-tile 0
            float h00 = acc00[r] + bc0;
            int y00 = (int)(h00 * 1565.0f + 16384.0f);  // trunc toward zero
            y00 >>= 15;                                  // arithmetic shift
            hbuf[wave][m][c0] = fminf(fmaxf((float)y00, 0.f), 255.f);
            // M-tile 1
            float h10 = acc10[r] + bc0;
            int y10 = (int)(h10 * 1565.0f + 16384.0f);
            y10 >>= 15;
            hbuf[wave][16 + m][c0] = fminf(fmaxf((float)y10, 0.f), 255.f);
            if (c1 < N1) {
                float h01 = acc01[r] + bc1;
                int y01 = (int)(h01 * 1565.0f + 16384.0f);
                y01 >>= 15;
                hbuf[wave][m][c1] = fminf(fmaxf((float)y01, 0.f), 255.f);
                float h11 = acc11[r] + bc1;
                int y11 = (int)(h11 * 1565.0f + 16384.0f);
                y11 >>= 15;
                hbuf[wave][16 + m][c1] = fminf(fmaxf((float)y11, 0.f), 255.f);
            }
        }
        __syncthreads();

        // ---- layer 2: [32,20] x [20,2]; each lane handles 2 rows ----
        {
            const int j = lane & 1;
            const float bj = b2[j];
            #pragma unroll
            for (int half = 0; half < 2; ++half) {
                const int r = (lane >> 1) + half * 16;
                float s = bj;
                #pragma unroll
                for (int i = 0; i < N1; ++i)
                    s += hbuf[wave][r][i] * W2[j * N1 + i];
                int y = (int)(s * 1342.0f + 16384.0f);
                y >>= 15;
                const float o = fminf(fmaxf((float)y, 0.f), 255.f);
                const int rowO = rowBase + wave * 32 + r;
                out[(size_t)rowO * 2 + j] = o;           // coalesced per wave
            }
        }
        __syncthreads();  // protect hbuf before next tile overwrites it
    }
}

extern "C" void kernel_launch(void* const* d_in, const int* in_sizes, int n_in,
                              void* d_out, int out_size, void* d_ws, size_t ws_size,
                              hipStream_t stream) {
    const float* X  = (const float*)d_in[0];
    const float* W1 = (const float*)d_in[1];
    const float* b1 = (const float*)d_in[2];
    const float* W2 = (const float*)d_in[3];
    const float* b2 = (const float*)d_in[4];
    float* out = (float*)d_out;

    const int nRows  = in_sizes[0] / K1;
    const int nTiles = nRows / ROWS_PER_BLOCK;    // B=1048576 -> 4096 tiles
    int blocks = nTiles < 2048 ? (nTiles > 0 ? nTiles : 1) : 2048;

    hipLaunchKernelGGL(fann_mlp_wmma_kernel, dim3(blocks), dim3(THREADS), 0, stream,
                       X, W1, b1, W2, b2, out, nTiles);
}